// TMessagePassing_11974368821731
// MI455X (gfx1250) — compile-verified
//
#include <hip/hip_runtime.h>
#include <hip/hip_bf16.h>
#include <cstdint>

// Hypergraph message passing: out[v,:] = coef * (2*x[v]*S1[v] + S2[v]),
//   S1[v] = sum_k x[nbr[v,k]],  S2[v] = sum_k x[nbr[v,k]]^2,  coef = (2/6)/K.
// N=100000 nodes, K=16 neighbors, D=128 features. x (51.2 MB) is L2-resident
// on MI455X (192 MB L2); kernel is L2-gather-bandwidth bound. WMMA is
// inapplicable (elementwise gather-reduce, no shared contraction dim), so the
// gfx1250-specific path used is the async global->LDS pipe for index staging.

typedef float v4f __attribute__((ext_vector_type(4)));

#define KDEG  16
#define DFEAT 128
#define ROW4  (DFEAT / 4)   // 32 float4 per feature row -> one row per wave32 load
#define NPB   8             // nodes per block = waves per block (wave32)

#if defined(__HIP_DEVICE_COMPILE__) && defined(__has_builtin)
#  if __has_builtin(__builtin_amdgcn_global_load_async_to_lds_b32)
#    define USE_ASYNC_LDS 1
#  endif
#endif

__global__ __launch_bounds__(NPB * 32) void
hyper_mp_kernel(const float* __restrict__ x,
                const int*   __restrict__ nbr,
                float*       __restrict__ out,
                int n)
{
    __shared__ int s_idx[NPB * KDEG];   // 128 indices for this block's 8 nodes

    const int tid   = (int)threadIdx.x;
    const int lane  = tid & 31;
    const int wave  = tid >> 5;
    const int vbase = (int)blockIdx.x * NPB;

    // ---- Stage this block's neighbor indices into LDS (async on gfx1250) ----
    const int nidx = min(NPB * KDEG, (n - vbase) * KDEG);
#if USE_ASYNC_LDS
    if (tid < nidx) {
        __builtin_amdgcn_global_load_async_to_lds_b32(
            (__attribute__((address_space(1))) int*)(nbr + (size_t)vbase * KDEG + tid),
            (__attribute__((address_space(3))) int*)(s_idx + tid),
            /*offset=*/0, /*cpol=*/0);
    }
#  if __has_builtin(__builtin_amdgcn_s_wait_asynccnt)
    __builtin_amdgcn_s_wait_asynccnt(0);
#  else
    asm volatile("s_wait_asynccnt 0" ::: "memory");
#  endif
    __syncthreads();
#else
    if (tid < nidx) s_idx[tid] = nbr[(size_t)vbase * KDEG + tid];
    __syncthreads();
#endif

    // ---- One wave per node; each lane owns 4 consecutive features ----
    const int v = vbase + wave;
    if (v >= n) return;

    const v4f* __restrict__ x4 = (const v4f*)x;
    const v4f  xv = x4[v * ROW4 + lane];           // global_load_b128, coalesced

    v4f s1 = {0.f, 0.f, 0.f, 0.f};
    v4f s2 = {0.f, 0.f, 0.f, 0.f};
    const int* widx = s_idx + wave * KDEG;

#pragma unroll
    for (int k = 0; k < KDEG; ++k) {
        const int u  = widx[k];                    // ds_load_b32 broadcast
        const v4f xu = x4[u * ROW4 + lane];        // 512B/wave gather, L2-resident
        s1 += xu;                                  // v_add_f32
        s2 += xu * xu;                             // v_fmac_f32
    }

    const float coef = (2.0f / 6.0f) / (float)KDEG;   // 1/48
    v4f o = (xv * s1 * 2.0f + s2) * coef;

    // Non-temporal store: don't evict the L2-resident x with output data.
    __builtin_nontemporal_store(o, (v4f*)out + v * ROW4 + lane);
}

extern "C" void kernel_launch(void* const* d_in, const int* in_sizes, int n_in,
                              void* d_out, int out_size, void* d_ws, size_t ws_size,
                              hipStream_t stream) {
    const float* x   = (const float*)d_in[0];
    const int*   nbr = (const int*)d_in[1];
    float*       out = (float*)d_out;

    const int N = in_sizes[0] / DFEAT;             // 100000
    dim3 block(NPB * 32);                          // 256 threads = 8 wave32
    dim3 grid((N + NPB - 1) / NPB);                // 12500 blocks

    hipLaunchKernelGGL(hyper_mp_kernel, grid, block, 0, stream, x, nbr, out, N);
}